// RSMNet_34909494182140
// MI455X (gfx1250) — compile-verified
//
#include <hip/hip_runtime.h>

// ---------------------------------------------------------------------------
// RSM-like network forward on MI455X (gfx1250).
// GEMMs: V_WMMA_F32_16X16X4_F32 (exact fp32 — the problem is L2/HBM bound,
// W_b (144MB) fits the 192MB L2, so fp32 WMMA loses nothing vs bf16) with the
// shared B tile staged through LDS by the Tensor Data Mover (double-buffered,
// TENSORcnt-pipelined), and A fragments preloaded in register chunks.
// ---------------------------------------------------------------------------

typedef float v2f __attribute__((ext_vector_type(2)));
typedef float v8f __attribute__((ext_vector_type(8)));
typedef unsigned int v4u __attribute__((ext_vector_type(4)));
typedef int v4i __attribute__((ext_vector_type(4)));
typedef int v8i __attribute__((ext_vector_type(8)));

#define B_ROWS   512
#define M_COLS   1000
#define N_CELLS  6
#define TC_      (M_COLS * N_CELLS)   // 6000
#define D_IN_    2048
#define D_OUT_   2048
#define K_TOP    50
#define GAMMA_   0.5f
#define NEG_BIG  (-3.402823466e38f)

#define KB          64                 // K-slab per TDM tile
#define LDS_STRIDE  68                 // 64 + 4 pad dwords -> conflict-free b64 reads

__device__ __forceinline__ unsigned map_f32(float f) {
  // order-preserving float -> uint map (larger float <=> larger uint)
  unsigned u = __float_as_uint(f);
  return (u & 0x80000000u) ? ~u : (u | 0x80000000u);
}

// ---------------------------------------------------------------------------
// Issue one TDM 2D tile load: W[n0 .. n0+63, kb .. kb+KB) -> LDS (row-major,
// row stride LDS_STRIDE floats via TDM pad). Tensor dims are the REMAINING
// extent from the tile start so edge tiles / K-tails are zero-filled by HW.
// ---------------------------------------------------------------------------
__device__ __forceinline__ void tdm_issue_b(const float* W, int n0, int kb,
                                            int Ncols, int Kdim, unsigned lds_off)
{
  unsigned long long ga = (unsigned long long)(size_t)(W + (size_t)n0 * Kdim + kb);
  unsigned td0 = (unsigned)(Kdim - kb);    // remaining K elements
  unsigned td1 = (unsigned)(Ncols - n0);   // remaining rows
  unsigned st0 = (unsigned)Kdim;           // row stride (elements)

  // D# group 0: count=1 | lds_addr | global_addr | type=2 ("image")
  v4u g0 = { 1u,
             lds_off,
             (unsigned)(ga & 0xFFFFFFFFull),
             (unsigned)((ga >> 32) & 0x01FFFFFFull) | (2u << 30) };

  // D# group 1:
  //  w0: data_size=4B(2<<16) | pad_enable(1<<20) | pad_interval=64DW(5<<22)
  //      | pad_amount=4DW(3<<25)
  //  w1[31:16]=tensor_dim0[15:0]         w2[15:0]=tensor_dim0[31:16]
  //  w2[31:16]=tensor_dim1[15:0]         w3[15:0]=tensor_dim1[31:16]
  //  w3[31:16]=tile_dim0=KB              w4[15:0]=tile_dim1=64, w4[31:16]=tile_dim2=0
  //  w5=tensor_dim0_stride[31:0]         w6[15:0]=stride[47:32]=0
  //  w6[31:16]=tensor_dim1_stride[15:0]=0 (unused, 2D tile)   w7=0
  v8i g1 = { (int)((2u << 16) | (1u << 20) | (5u << 22) | (3u << 25)),
             (int)((td0 & 0xFFFFu) << 16),
             (int)(((td0 >> 16) & 0xFFFFu) | ((td1 & 0xFFFFu) << 16)),
             (int)(((td1 >> 16) & 0xFFFFu) | ((unsigned)KB << 16)),
             (int)64,
             (int)st0,
             0, 0 };

  v4i z4 = { 0, 0, 0, 0 };  // groups 2/3 unused (2D tile: tile_dim2/3/4 = 0)

#if __has_include(<hip/amd_detail/amd_gfx1250_TDM.h>)
  v8i z8 = { 0, 0, 0, 0, 0, 0, 0, 0 };
  __builtin_amdgcn_tensor_load_to_lds(g0, g1, z4, z4, z8, 0);
#else
  __builtin_amdgcn_tensor_load_to_lds(g0, g1, z4, z4, 0);
#endif
}

// ---------------------------------------------------------------------------
// C[M,N] = A[M,K] * W[N,K]^T  (+ bias[N]) (+ zrep[row, n/N_CELLS])
// block = 256 thr = 8 waves; block tile 128 rows x 64 cols; wave tile 16x64.
// B slab TDM-staged into LDS (double-buffered); A chunk preloaded to regs.
// A frag: row = lane&15, k-off = (lane>>4)*2 -> float2
// B frag: col = lane&15, k-off = (lane>>4)*2 -> float2 (from LDS)
// D frag: row = (lane>>4)*8 + v, col = lane&15
// ---------------------------------------------------------------------------
template <bool ADD_BIAS, bool ADD_REPEAT>
__global__ __launch_bounds__(256)
void gemm_wmma_tdm(const float* __restrict__ A, const float* __restrict__ W,
                   const float* __restrict__ bias, const float* __restrict__ zrep,
                   float* __restrict__ C, int Ncols, int Kdim, int zcols)
{
  __shared__ float bsm[2][64 * LDS_STRIDE];   // 2 x 17408 B

  const int lane = threadIdx.x & 31;
  const int wave = threadIdx.x >> 5;
  const int m0   = blockIdx.y * 128 + wave * 16;
  const int n0   = blockIdx.x * 64;
  const int l15  = lane & 15;
  const int k0   = (lane >> 4) * 2;

  const float* Ar = A + (size_t)(m0 + l15) * Kdim;
  const int nslab = (Kdim + KB - 1) / KB;

  const unsigned lds0 = (unsigned)(size_t)&bsm[0][0];  // low 32 bits = LDS offset
  const unsigned lds1 = (unsigned)(size_t)&bsm[1][0];

  if (wave == 0) tdm_issue_b(W, n0, 0, Ncols, Kdim, lds0);

  v8f acc[4];
#pragma unroll
  for (int t = 0; t < 4; ++t) acc[t] = (v8f){0.f,0.f,0.f,0.f,0.f,0.f,0.f,0.f};

  for (int s = 0; s < nslab; ++s) {
    const int kb = s * KB;
    if (wave == 0) {
      if (s + 1 < nslab) {
        tdm_issue_b(W, n0, kb + KB, Ncols, Kdim, (s & 1) ? lds0 : lds1);
        __builtin_amdgcn_s_wait_tensorcnt(1);   // slab s complete (in-order)
      } else {
        __builtin_amdgcn_s_wait_tensorcnt(0);
      }
    }
    __syncthreads();   // slab s visible to all waves

    const float* bufc = &bsm[s & 1][0];

    // preload this slab's A fragments (16 x float2) so loads batch up front
    v2f areg[16];
    if (kb + KB <= Kdim) {
#pragma unroll
      for (int i = 0; i < 16; ++i)
        areg[i] = *(const v2f*)(Ar + kb + i * 4 + k0);
    } else {
#pragma unroll
      for (int i = 0; i < 16; ++i) {            // K-tail: guarded loads
        int k = kb + i * 4 + k0;
        v2f t;
        t.x = (k     < Kdim) ? Ar[k]     : 0.0f;
        t.y = (k + 1 < Kdim) ? Ar[k + 1] : 0.0f;
        areg[i] = t;
      }
    }

#pragma unroll
    for (int i = 0; i < 16; ++i) {
      const int koff = i * 4 + k0;
      v2f b0 = *(const v2f*)(bufc + (0 * 16 + l15) * LDS_STRIDE + koff);
      v2f b1 = *(const v2f*)(bufc + (1 * 16 + l15) * LDS_STRIDE + koff);
      v2f b2 = *(const v2f*)(bufc + (2 * 16 + l15) * LDS_STRIDE + koff);
      v2f b3 = *(const v2f*)(bufc + (3 * 16 + l15) * LDS_STRIDE + koff);
      acc[0] = __builtin_amdgcn_wmma_f32_16x16x4_f32(false, areg[i], false, b0, (short)0, acc[0], false, false);
      acc[1] = __builtin_amdgcn_wmma_f32_16x16x4_f32(false, areg[i], false, b1, (short)0, acc[1], false, false);
      acc[2] = __builtin_amdgcn_wmma_f32_16x16x4_f32(false, areg[i], false, b2, (short)0, acc[2], false, false);
      acc[3] = __builtin_amdgcn_wmma_f32_16x16x4_f32(false, areg[i], false, b3, (short)0, acc[3], false, false);
    }
    __syncthreads();   // all waves done reading buf[s&1] before TDM reuses it
  }

  const int mrow = m0 + (lane >> 4) * 8;
#pragma unroll
  for (int t = 0; t < 4; ++t) {
    int n = n0 + t * 16 + l15;
    if (n < Ncols) {
      float bval = ADD_BIAS ? bias[n] : 0.0f;
      int   zc   = ADD_REPEAT ? (n / N_CELLS) : 0;
#pragma unroll
      for (int v = 0; v < 8; ++v) {
        int   r   = mrow + v;
        float val = acc[t][v] + bval;
        if (ADD_REPEAT) val += zrep[(size_t)r * zcols + zc];
        C[(size_t)r * Ncols + n] = val;
      }
    }
  }
}

// ---------------------------------------------------------------------------
// per-row min of sigma[b, 0..TC)
// ---------------------------------------------------------------------------
__global__ __launch_bounds__(256)
void rowmin_kernel(const float* __restrict__ sigma, float* __restrict__ rmin)
{
  const int b = blockIdx.x;
  __shared__ float red[256];
  float mn = 3.402823466e38f;
  for (int i = threadIdx.x; i < TC_; i += 256)
    mn = fminf(mn, sigma[(size_t)b * TC_ + i]);
  red[threadIdx.x] = mn;
  __syncthreads();
  for (int s = 128; s > 0; s >>= 1) {
    if (threadIdx.x < s) red[threadIdx.x] = fminf(red[threadIdx.x], red[threadIdx.x + s]);
    __syncthreads();
  }
  if (threadIdx.x == 0) rmin[b] = red[0];
}

// ---------------------------------------------------------------------------
// lam[b,m] = max_i (1-phi)*(sigma - rowmin)   (stored as mapped uint)
// ---------------------------------------------------------------------------
__global__ __launch_bounds__(256)
void lam_kernel(const float* __restrict__ sigma, const float* __restrict__ phi,
                const float* __restrict__ rmin, unsigned* __restrict__ lam_u)
{
  int idx = blockIdx.x * 256 + threadIdx.x;
  if (idx >= B_ROWS * M_COLS) return;
  int b = idx / M_COLS, m = idx - b * M_COLS;
  float rm = rmin[b];
  size_t base = (size_t)b * TC_ + (size_t)m * N_CELLS;
  float lam = NEG_BIG;
#pragma unroll
  for (int i = 0; i < N_CELLS; ++i) {
    float pi = (1.0f - phi[base + i]) * (sigma[base + i] - rm);
    lam = fmaxf(lam, pi);
  }
  lam_u[idx] = map_f32(lam);
}

// ---------------------------------------------------------------------------
// exact k-th largest per row via 32-bit radix select (one block per row)
// ---------------------------------------------------------------------------
__global__ __launch_bounds__(256)
void select_thr_kernel(const unsigned* __restrict__ lam_u, unsigned* __restrict__ thr)
{
  const int b = blockIdx.x;
  __shared__ unsigned keys[1024];
  __shared__ int s_cnt;
  for (int i = threadIdx.x; i < M_COLS; i += 256)
    keys[i] = lam_u[(size_t)b * M_COLS + i];
  __syncthreads();

  unsigned prefix = 0;
  int kk = K_TOP;
  for (int bit = 31; bit >= 0; --bit) {
    if (threadIdx.x == 0) s_cnt = 0;
    __syncthreads();
    unsigned want = (prefix >> bit) | 1u;
    int c = 0;
    for (int i = threadIdx.x; i < M_COLS; i += 256)
      c += ((keys[i] >> bit) == want) ? 1 : 0;
    atomicAdd(&s_cnt, c);
    __syncthreads();
    int ctot = s_cnt;
    if (ctot >= kk) prefix |= (1u << bit);
    else            kk -= ctot;
    __syncthreads();
  }
  if (threadIdx.x == 0) thr[b] = prefix;  // mapped-uint of 50th largest lam
}

// ---------------------------------------------------------------------------
// masks + tanh + decayed max updates; also column-max of psi_n for decode
// ---------------------------------------------------------------------------
__global__ __launch_bounds__(256)
void mask_update_kernel(const float* __restrict__ sigma, const float* __restrict__ phi,
                        const float* __restrict__ psi, const float* __restrict__ decay,
                        const float* __restrict__ rmin, const unsigned* __restrict__ lam_u,
                        const unsigned* __restrict__ thr,
                        float* __restrict__ xbn, float* __restrict__ phin,
                        float* __restrict__ psin, float* __restrict__ ucol)
{
  int idx = blockIdx.x * 256 + threadIdx.x;
  if (idx >= B_ROWS * M_COLS) return;
  int b = idx / M_COLS, m = idx - b * M_COLS;
  float rm = rmin[b];
  size_t base  = (size_t)b * TC_ + (size_t)m * N_CELLS;
  size_t dbase = (size_t)m * N_CELLS;

  // argmax of pi within the column; first-index tie-break (== lax.top_k)
  float best = NEG_BIG;
  int   j    = 0;
#pragma unroll
  for (int i = 0; i < N_CELLS; ++i) {
    float pi = (1.0f - phi[base + i]) * (sigma[base + i] - rm);
    if (pi > best) { best = pi; j = i; }
  }
  bool col_win = (lam_u[idx] >= thr[b]);  // mapped-space compare == lam >= thr

  float umax = NEG_BIG;
#pragma unroll
  for (int i = 0; i < N_CELLS; ++i) {
    float y  = (col_win && i == j) ? tanhf(sigma[base + i]) : 0.0f;
    float pn = fmaxf(psi[base + i] * decay[dbase + i], y);
    float fn = fmaxf(phi[base + i] * GAMMA_, y);
    psin[base + i] = pn;
    phin[base + i] = fn;
    xbn[base + i]  = pn;  // x_b_norm = False -> x_b_n == psi_n
    umax = fmaxf(umax, pn);
  }
  ucol[idx] = umax;
}

// ---------------------------------------------------------------------------
extern "C" void kernel_launch(void* const* d_in, const int* in_sizes, int n_in,
                              void* d_out, int out_size, void* d_ws, size_t ws_size,
                              hipStream_t stream)
{
  (void)in_sizes; (void)n_in; (void)out_size; (void)ws_size;

  const float* x_a   = (const float*)d_in[0];
  const float* x_b   = (const float*)d_in[1];
  const float* phi   = (const float*)d_in[2];
  const float* psi   = (const float*)d_in[3];
  const float* W_a   = (const float*)d_in[4];
  const float* W_b   = (const float*)d_in[5];
  const float* W_d   = (const float*)d_in[6];
  const float* b_d   = (const float*)d_in[7];
  const float* decay = (const float*)d_in[8];

  float* out  = (float*)d_out;                         // [512, 2048]
  float* xbn  = out  + (size_t)B_ROWS * D_OUT_;        // [512, 6000]
  float* phin = xbn  + (size_t)B_ROWS * TC_;           // [512, 6000]
  float* psin = phin + (size_t)B_ROWS * TC_;           // [512, 6000]

  float*    ws    = (float*)d_ws;
  float*    z_a   = ws;                                     // 512*1000
  float*    sigma = z_a + (size_t)B_ROWS * M_COLS;          // 512*6000
  float*    rmin  = sigma + (size_t)B_ROWS * TC_;           // 512
  unsigned* lam_u = (unsigned*)(rmin + B_ROWS);             // 512*1000
  unsigned* thr   = lam_u + (size_t)B_ROWS * M_COLS;        // 512
  float*    ucol  = (float*)(thr + B_ROWS);                 // 512*1000
  // total scratch ~ 18.5 MB

  dim3 blk(256);

  // 1) z_a = x_a @ W_a^T                       [512,1000] K=2048
  gemm_wmma_tdm<false, false>
      <<<dim3((M_COLS + 63) / 64, B_ROWS / 128), blk, 0, stream>>>(
          x_a, W_a, nullptr, nullptr, z_a, M_COLS, D_IN_, 0);

  // 2) sigma = x_b @ W_b^T + repeat(z_a, 6)    [512,6000] K=6000
  gemm_wmma_tdm<false, true>
      <<<dim3((TC_ + 63) / 64, B_ROWS / 128), blk, 0, stream>>>(
          x_b, W_b, nullptr, z_a, sigma, TC_, TC_, M_COLS);

  // 3) row minima of sigma
  rowmin_kernel<<<B_ROWS, blk, 0, stream>>>(sigma, rmin);

  // 4) column activities lam (mapped uints)
  lam_kernel<<<(B_ROWS * M_COLS + 255) / 256, blk, 0, stream>>>(sigma, phi, rmin, lam_u);

  // 5) exact top-50 threshold per row (radix select)
  select_thr_kernel<<<B_ROWS, blk, 0, stream>>>(lam_u, thr);

  // 6) masks, tanh, decayed-max state updates, column max of psi_n
  mask_update_kernel<<<(B_ROWS * M_COLS + 255) / 256, blk, 0, stream>>>(
      sigma, phi, psi, decay, rmin, lam_u, thr, xbn, phin, psin, ucol);

  // 7) out = ucol @ W_d^T + b_d                [512,2048] K=1000
  gemm_wmma_tdm<true, false>
      <<<dim3((D_OUT_ + 63) / 64, B_ROWS / 128), blk, 0, stream>>>(
          ucol, W_d, b_d, nullptr, out, D_OUT_, M_COLS, 0);
}